// eDCC_loss_85349590106761
// MI455X (gfx1250) — compile-verified
//
#include <hip/hip_runtime.h>
#include <math.h>

// ---------------------------------------------------------------------------
// eDCC loss: only planes i=127 and j=29 of projs[1,128,1024,256] matter.
// P_i[l] = sum_s projs[127,l,s] * exp(alpha*(s-127.5)),  alpha = sigma*SPACING^2
// term[l] = den!=0 ? (2/128)*|P_i-P_j|/den : 0,  den = P_i+P_j
// out = sum_l term[l]
//
// Dot products are executed on the matrix pipe with V_WMMA_F32_16X16X4_F32
// (f32 operands are mandatory: exp weights span ~1e-16..1e15, far beyond
// f16/bf16/fp8 range). Memory-bound (~2 MB total); WMMA is the compute path.
// ---------------------------------------------------------------------------

typedef float v2f __attribute__((ext_vector_type(2)));
typedef float v8f __attribute__((ext_vector_type(8)));

#define NPROJS 128
#define LROWS  1024
#define SCOLS  256

__global__ __launch_bounds__(32)
void edcc_wmma_kernel(const float* __restrict__ projs,
                      float* __restrict__ partials,   // [64] per-tile sums
                      float alpha)
{
    const int tile = blockIdx.x;           // 0..63 -> 16 rows each
    const int lane = threadIdx.x;          // 0..31 (wave32)
    const int kh   = lane >> 4;            // 0: K0/K1 ; 1: K2/K3  (A layout)
    const int row  = (tile << 4) + (lane & 15);

    const float* rowI = projs + (size_t)127 * LROWS * SCOLS + (size_t)row * SCOLS;
    const float* rowJ = projs + (size_t) 29 * LROWS * SCOLS + (size_t)row * SCOLS;

    // B weights: w(k) = exp(alpha*(k - 127.5)); advanced by running product.
    const float k0f  = (float)(2 * kh) - 127.5f;
    float w0 = expf(alpha * k0f);
    float w1 = expf(alpha * (k0f + 1.0f));
    const float step = expf(4.0f * alpha);

    v8f accI = {};
    v8f accJ = {};

    for (int c = 0; c < SCOLS / 4; ++c) {
        const int kb = 4 * c + 2 * kh;
        // 32-bit A-matrix 16x4 layout: lanes 0-15 -> {K=0,K=1}, lanes 16-31 -> {K=2,K=3}
        v2f ai = *(const v2f*)(rowI + kb);
        v2f aj = *(const v2f*)(rowJ + kb);
        // B 4x16: all 16 columns get the same weight chunk (broadcast).
        v2f b;
        b.x = w0;
        b.y = w1;
        // D = A*B + C  (8 args: neg_a, A, neg_b, B, c_mod, C, reuse_a, reuse_b)
        accI = __builtin_amdgcn_wmma_f32_16x16x4_f32(
                   false, ai, false, b, (short)0, accI, false, false);
        accJ = __builtin_amdgcn_wmma_f32_16x16x4_f32(
                   false, aj, false, b, (short)0, accJ, false, false);
        w0 *= step;
        w1 *= step;
    }

    // C/D layout: lanes 0-15 hold rows tile*16 + v (v=0..7) in accX[v];
    // lanes 16-31 hold rows tile*16 + 8 + v. Every column is identical.
    float local = 0.0f;
    #pragma unroll
    for (int v = 0; v < 8; ++v) {
        float pi  = accI[v];
        float pj  = accJ[v];
        float den = pi + pj;
        float t   = (den != 0.0f) ? (2.0f / NPROJS) * fabsf(pi - pj) / den : 0.0f;
        local += t;
    }
    // Combine lower-half (rows 0-7) and upper-half (rows 8-15) contributions.
    float other = __shfl(local, 16, 32);   // lane 0 reads lane 16's value
    if (lane == 0)
        partials[tile] = local + other;
}

__global__ __launch_bounds__(32)
void edcc_reduce_kernel(const float* __restrict__ partials,
                        float* __restrict__ out)
{
    // Single-thread fixed-order reduction: bitwise deterministic across calls.
    if (threadIdx.x == 0 && blockIdx.x == 0) {
        float s = 0.0f;
        for (int i = 0; i < 64; ++i)
            s += partials[i];
        out[0] = s;   // B == 1, so mean over batch == the sum itself
    }
}

extern "C" void kernel_launch(void* const* d_in, const int* in_sizes, int n_in,
                              void* d_out, int out_size, void* d_ws, size_t ws_size,
                              hipStream_t stream)
{
    (void)in_sizes; (void)n_in; (void)out_size; (void)ws_size;

    const float* projs    = (const float*)d_in[0];
    float*       out      = (float*)d_out;
    float*       partials = (float*)d_ws;      // 64 floats of scratch

    // Host-side constants (deterministic, pure): theta step = 2*pi/128.
    const double twoPi  = 6.28318530717958647692;
    const double dtheta = twoPi * (127.0 - 29.0) / 128.0;     // theta[i]-theta[j]
    const double sigma  = 0.013 * tan(dtheta / 2.0);
    const double sp     = 4.7952;
    const float  alpha  = (float)(sigma * sp * sp);           // exp(alpha*(k-127.5))

    edcc_wmma_kernel<<<64, 32, 0, stream>>>(projs, partials, alpha);
    edcc_reduce_kernel<<<1, 32, 0, stream>>>(partials, out);
}